// GraphAttention_59339268161756
// MI455X (gfx1250) — compile-verified
//
#include <hip/hip_runtime.h>
#include <hip/hip_bf16.h>

// GAT layer for MI455X (gfx1250, wave32).
// B=8, N=2048, D=128, H=4, d=32. NEG_SLOPE=0.2
// Outputs: out (B,N,D) then attn (B,H,N,N), concatenated in d_out.
//
// Traffic plan (23.3 TB/s HBM, 192 MB L2):
//   streaming (NT hints): adj 2x134MB, attn write 536MB
//   L2-resident: Wh (8 MB) re-read ~128x by k_attn
//   ~0.85 GB HBM  =>  ~37 us floor; all GEMMs via v_wmma_f32_16x16x4_f32.

typedef float v2f __attribute__((ext_vector_type(2)));
typedef float v8f __attribute__((ext_vector_type(8)));
typedef int v4i __attribute__((ext_vector_type(4)));

#define GAT_B 8
#define GAT_N 2048
#define GAT_D 128
#define GAT_H 4
#define GAT_d 32
#define NEG_SLOPE 0.2f

#define V8F_ZERO {0.f, 0.f, 0.f, 0.f, 0.f, 0.f, 0.f, 0.f}

// ---------------------------------------------------------------------------
// Kernel 1: Wh = x @ W_lin + b_lin  (16384x128 @ 128x128), stored [b][h][n][d]
// One wave per 16x16 output tile; 32 x v_wmma_f32_16x16x4_f32 on 2 chains.
// grid (1024, 2), block 128 (4 waves)
// ---------------------------------------------------------------------------
__global__ void k_lin(const float* __restrict__ x, const float* __restrict__ W,
                      const float* __restrict__ bias, float* __restrict__ Wh) {
  const int wave = threadIdx.x >> 5;
  const int lane = threadIdx.x & 31;
  const int m0 = blockIdx.x * 16;
  const int n0 = (blockIdx.y * 4 + wave) * 16;
  const int half = lane >> 4;  // 0: K+{0,1}, 1: K+{2,3}
  const int l = lane & 15;

  v8f c0 = V8F_ZERO, c1 = V8F_ZERO;
  const float* xrow = x + (size_t)(m0 + l) * GAT_D;
#pragma unroll 2
  for (int k0 = 0; k0 < GAT_D; k0 += 8) {
    v2f av, bv;
    int k = k0 + half * 2;
    av.x = xrow[k];
    av.y = xrow[k + 1];
    bv.x = W[(size_t)k * GAT_D + n0 + l];
    bv.y = W[(size_t)(k + 1) * GAT_D + n0 + l];
    c0 = __builtin_amdgcn_wmma_f32_16x16x4_f32(false, av, false, bv, (short)0,
                                               c0, false, false);
    k = k0 + 4 + half * 2;
    av.x = xrow[k];
    av.y = xrow[k + 1];
    bv.x = W[(size_t)k * GAT_D + n0 + l];
    bv.y = W[(size_t)(k + 1) * GAT_D + n0 + l];
    c1 = __builtin_amdgcn_wmma_f32_16x16x4_f32(false, av, false, bv, (short)0,
                                               c1, false, false);
  }
  const v8f c = c0 + c1;
  const int col = n0 + l;
  const float bb = bias[col];
  const int h = col >> 5, dd = col & 31;
#pragma unroll
  for (int v = 0; v < 8; ++v) {
    const int r = m0 + v + half * 8;  // global row = b*N + n
    const int b_ = r >> 11;
    const int n = r & (GAT_N - 1);
    Wh[(((size_t)(b_ * GAT_H + h)) * GAT_N + n) * GAT_d + dd] = c[v] + bb;
  }
}

// ---------------------------------------------------------------------------
// Kernel 2: h_left[b][h][n] = Wh row . a[h][:32]; h_right = Wh row . a[h][32:]
// ---------------------------------------------------------------------------
__global__ void k_hl(const float* __restrict__ Wh, const float* __restrict__ a,
                     float* __restrict__ hl, float* __restrict__ hr) {
  const int t = blockIdx.x * blockDim.x + threadIdx.x;  // (b*H+h)*N + n
  if (t >= GAT_B * GAT_H * GAT_N) return;
  const int h = (t >> 11) & (GAT_H - 1);
  const float* row = Wh + (size_t)t * GAT_d;
  const float* al = a + h * 2 * GAT_d;
  const float* ar = al + GAT_d;
  float sl = 0.f, sr = 0.f;
#pragma unroll
  for (int k = 0; k < GAT_d; ++k) {
    sl += row[k] * al[k];
    sr += row[k] * ar[k];
  }
  hl[t] = sl;
  hr[t] = sr;
}

// ---------------------------------------------------------------------------
// Kernel 3: online-softmax stats (row max, row sum) for all 4 heads at once,
// reading adj_mask once per (b,i,j) (non-temporal: pure streaming traffic).
// h_right staged in 32KB LDS via async-LDS copies when the toolchain has them.
// grid (N/16, B), block 256 (8 waves, 2 rows per wave).
// ---------------------------------------------------------------------------
__global__ void k_stats(const int* __restrict__ adj, const float* __restrict__ hl,
                        const float* __restrict__ hr, float* __restrict__ rmax,
                        float* __restrict__ rsum) {
  const int b = blockIdx.y;
  const int i0 = blockIdx.x * 16;
  __shared__ float hr_s[GAT_H][GAT_N];
  {
    const float* src = hr + ((size_t)(b * GAT_H) << 11);
    float* hs = &hr_s[0][0];
#if __has_builtin(__builtin_amdgcn_global_load_async_to_lds_b128)
    for (int idx = threadIdx.x * 4; idx < GAT_H * GAT_N; idx += blockDim.x * 4) {
      __builtin_amdgcn_global_load_async_to_lds_b128((v4i*)(src + idx),
                                                     (v4i*)(hs + idx), 0, 0);
    }
#if __has_builtin(__builtin_amdgcn_s_wait_asynccnt)
    __builtin_amdgcn_s_wait_asynccnt(0);
#else
    asm volatile("s_wait_asynccnt 0x0" ::: "memory");
#endif
#else
    for (int idx = threadIdx.x; idx < GAT_H * GAT_N; idx += blockDim.x)
      hs[idx] = src[idx];
#endif
  }
  __syncthreads();

  const int wave = threadIdx.x >> 5;
  const int lane = threadIdx.x & 31;
  for (int rr = 0; rr < 2; ++rr) {
    const int i = i0 + wave * 2 + rr;
    float hlv[GAT_H];
#pragma unroll
    for (int h = 0; h < GAT_H; ++h) hlv[h] = hl[((b * GAT_H + h) << 11) + i];
    float m[GAT_H] = {-1e30f, -1e30f, -1e30f, -1e30f};
    float s[GAT_H] = {0.f, 0.f, 0.f, 0.f};
    const int* arow = adj + ((size_t)(b * GAT_N + i)) * GAT_N;
    for (int j = lane; j < GAT_N; j += 32) {
      const int aj = __builtin_nontemporal_load(arow + j);  // streaming
      if (aj != 0) {
#pragma unroll
        for (int h = 0; h < GAT_H; ++h) {
          float e = hlv[h] + hr_s[h][j];
          e = (e >= 0.f) ? e : NEG_SLOPE * e;
          if (e > m[h]) {
            s[h] = s[h] * __expf(m[h] - e) + 1.0f;
            m[h] = e;
          } else {
            s[h] += __expf(e - m[h]);
          }
        }
      }
    }
    // wave32 (m,s) tree merge
#pragma unroll
    for (int off = 16; off > 0; off >>= 1) {
#pragma unroll
      for (int h = 0; h < GAT_H; ++h) {
        const float mo = __shfl_xor(m[h], off, 32);
        const float so = __shfl_xor(s[h], off, 32);
        const float M = fmaxf(m[h], mo);
        s[h] = s[h] * __expf(m[h] - M) + so * __expf(mo - M);
        m[h] = M;
      }
    }
    if (lane == 0) {
#pragma unroll
      for (int h = 0; h < GAT_H; ++h) {
        rmax[((b * GAT_H + h) << 11) + i] = m[h];
        rsum[((b * GAT_H + h) << 11) + i] = s[h];
      }
    }
  }
}

// ---------------------------------------------------------------------------
// Kernel 4: fused attn write + attn @ Wh. One block per (b, 16-row i-tile);
// 4 waves = 4 heads. Per 16-wide j-tile: compute attn 16x16 in C-layout
// (coalesced NT adj load + NT attn store), bounce via a *wave-private* LDS
// tile into WMMA A-layout (no block barrier needed: per-wave DS ops are
// in-order), accumulate out 16x32 on 4 independent WMMA chains.
// grid (N/16, B), block 128.
// ---------------------------------------------------------------------------
__global__ void k_attn(const int* __restrict__ adj, const float* __restrict__ hl,
                       const float* __restrict__ hr, const float* __restrict__ rmax,
                       const float* __restrict__ rsum, const float* __restrict__ Wh,
                       float* __restrict__ attn, float* __restrict__ att_out) {
  const int b = blockIdx.y;
  const int i0 = blockIdx.x * 16;
  const int h = threadIdx.x >> 5;  // wave == head
  const int lane = threadIdx.x & 31;
  const int half = lane >> 4;
  const int l = lane & 15;
  const int roff = half * 8;

  __shared__ float p_s[GAT_H][16][17];  // per-wave private tile

  const int bh = b * GAT_H + h;
  float hlr[8], mr[8], isr[8];
#pragma unroll
  for (int v = 0; v < 8; ++v) {
    const int i = i0 + roff + v;
    hlr[v] = hl[(bh << 11) + i];
    mr[v] = rmax[(bh << 11) + i];
    const float s = rsum[(bh << 11) + i];
    isr[v] = (s > 0.f) ? (1.0f / s) : 0.f;
  }
  v8f c0a = V8F_ZERO, c0b = V8F_ZERO, c1a = V8F_ZERO, c1b = V8F_ZERO;

  const size_t adj_base = (size_t)(b * GAT_N + i0) * GAT_N;
  const size_t attn_base = ((size_t)bh * GAT_N + i0) * GAT_N;
  const float* WhB = Wh + (size_t)bh * GAT_N * GAT_d;

  for (int j0 = 0; j0 < GAT_N; j0 += 16) {
    // prefetch next j-tile's adjacency (global_prefetch_b8)
    if (j0 + 16 < GAT_N)
      __builtin_prefetch(adj + adj_base + (size_t)roff * GAT_N + j0 + 16 + l, 0, 1);

    const float hc = hr[(bh << 11) + j0 + l];
#pragma unroll
    for (int v = 0; v < 8; ++v) {
      const int row = roff + v;
      const int aj =
          __builtin_nontemporal_load(adj + adj_base + (size_t)row * GAT_N + j0 + l);
      float e = hlr[v] + hc;
      e = (e >= 0.f) ? e : NEG_SLOPE * e;
      const float p = aj ? __expf(e - mr[v]) * isr[v] : 0.f;
      // attn written exactly once, never re-read: keep it out of L2 (NT)
      __builtin_nontemporal_store(p, attn + attn_base + (size_t)row * GAT_N + j0 + l);
      p_s[h][row][l] = p;
    }
    __builtin_amdgcn_wave_barrier();  // order DS store -> DS load (same wave)
#pragma unroll
    for (int ks = 0; ks < 4; ++ks) {
      v2f av, b0, b1;
      const int k = ks * 4 + half * 2;  // A-layout K slot for this lane half
      av.x = p_s[h][l][k];
      av.y = p_s[h][l][k + 1];
      const float* w0 = WhB + (size_t)(j0 + k) * GAT_d;  // L2-resident
      b0.x = w0[l];
      b0.y = w0[GAT_d + l];
      b1.x = w0[16 + l];
      b1.y = w0[GAT_d + 16 + l];
      if (ks & 1) {
        c0b = __builtin_amdgcn_wmma_f32_16x16x4_f32(false, av, false, b0,
                                                    (short)0, c0b, false, false);
        c1b = __builtin_amdgcn_wmma_f32_16x16x4_f32(false, av, false, b1,
                                                    (short)0, c1b, false, false);
      } else {
        c0a = __builtin_amdgcn_wmma_f32_16x16x4_f32(false, av, false, b0,
                                                    (short)0, c0a, false, false);
        c1a = __builtin_amdgcn_wmma_f32_16x16x4_f32(false, av, false, b1,
                                                    (short)0, c1a, false, false);
      }
    }
    __builtin_amdgcn_wave_barrier();  // next iter overwrites the tile
  }
  const v8f c0 = c0a + c0b;
  const v8f c1 = c1a + c1b;
  // att_out laid out [b][n][h*32 + c] so the final projection is a plain GEMM
#pragma unroll
  for (int v = 0; v < 8; ++v) {
    const int n = i0 + roff + v;
    float* dst = att_out + ((size_t)(b * GAT_N + n)) * (GAT_H * GAT_d) + h * GAT_d;
    dst[l] = c0[v];
    dst[16 + l] = c1[v];
  }
}

// ---------------------------------------------------------------------------
// Kernel 5: out = att_out (16384x128) @ W_out (128x128) + b_out
// ---------------------------------------------------------------------------
__global__ void k_out(const float* __restrict__ xin, const float* __restrict__ W,
                      const float* __restrict__ bias, float* __restrict__ out) {
  const int wave = threadIdx.x >> 5;
  const int lane = threadIdx.x & 31;
  const int m0 = blockIdx.x * 16;
  const int n0 = (blockIdx.y * 4 + wave) * 16;
  const int half = lane >> 4;
  const int l = lane & 15;

  v8f c0 = V8F_ZERO, c1 = V8F_ZERO;
  const float* xrow = xin + (size_t)(m0 + l) * GAT_D;
#pragma unroll 2
  for (int k0 = 0; k0 < GAT_D; k0 += 8) {
    v2f av, bv;
    int k = k0 + half * 2;
    av.x = xrow[k];
    av.y = xrow[k + 1];
    bv.x = W[(size_t)k * GAT_D + n0 + l];
    bv.y = W[(size_t)(k + 1) * GAT_D + n0 + l];
    c0 = __builtin_amdgcn_wmma_f32_16x16x4_f32(false, av, false, bv, (short)0,
                                               c0, false, false);
    k = k0 + 4 + half * 2;
    av.x = xrow[k];
    av.y = xrow[k + 1];
    bv.x = W[(size_t)k * GAT_D + n0 + l];
    bv.y = W[(size_t)(k + 1) * GAT_D + n0 + l];
    c1 = __builtin_amdgcn_wmma_f32_16x16x4_f32(false, av, false, bv, (short)0,
                                               c1, false, false);
  }
  const v8f c = c0 + c1;
  const int col = n0 + l;
  const float bb = bias[col];
#pragma unroll
  for (int v = 0; v < 8; ++v) {
    const int r = m0 + v + half * 8;
    out[(size_t)r * GAT_D + col] = c[v] + bb;
  }
}

// ---------------------------------------------------------------------------
extern "C" void kernel_launch(void* const* d_in, const int* in_sizes, int n_in,
                              void* d_out, int out_size, void* d_ws, size_t ws_size,
                              hipStream_t stream) {
  const float* x = (const float*)d_in[0];
  const int* adj = (const int*)d_in[1];
  const float* W_lin = (const float*)d_in[2];
  const float* b_lin = (const float*)d_in[3];
  const float* a = (const float*)d_in[4];
  const float* W_out = (const float*)d_in[5];
  const float* b_out = (const float*)d_in[6];

  float* out = (float*)d_out;                         // (B,N,D)
  float* attn = out + (size_t)GAT_B * GAT_N * GAT_D;  // (B,H,N,N)

  float* ws = (float*)d_ws;
  float* Wh = ws;                                               // 2,097,152
  float* att_out = Wh + (size_t)GAT_B * GAT_H * GAT_N * GAT_d;  // 2,097,152
  float* hl = att_out + (size_t)GAT_B * GAT_N * GAT_D;          // 65,536
  float* hr = hl + GAT_B * GAT_H * GAT_N;                       // 65,536
  float* rmax = hr + GAT_B * GAT_H * GAT_N;                     // 65,536
  float* rsum = rmax + GAT_B * GAT_H * GAT_N;                   // 65,536

  k_lin<<<dim3(1024, 2), 128, 0, stream>>>(x, W_lin, b_lin, Wh);
  k_hl<<<dim3(256), 256, 0, stream>>>(Wh, a, hl, hr);
  k_stats<<<dim3(GAT_N / 16, GAT_B), 256, 0, stream>>>(adj, hl, hr, rmax, rsum);
  k_attn<<<dim3(GAT_N / 16, GAT_B), 128, 0, stream>>>(adj, hl, hr, rmax, rsum,
                                                      Wh, attn, att_out);
  k_out<<<dim3(1024, 2), 128, 0, stream>>>(att_out, W_out, b_out, out);
}